// S3Loss_9165460210054
// MI455X (gfx1250) — compile-verified
//
#include <hip/hip_runtime.h>
#include <hip/hip_bf16.h>
#include <stdint.h>

typedef float v2f __attribute__((ext_vector_type(2)));
typedef float v8f __attribute__((ext_vector_type(8)));

#define B_   4
#define C_   8
#define H_   1024
#define W_   1024
#define HW_  (H_*W_)
#define hs_  256
#define ws2_ 256
#define N_   (hs_*ws2_)     // 65536 low-res pixels per batch
#define KBLUR 31
#define PADB  15
#define EPS_  1e-10f

// ---- workspace layout (float offsets) ----
#define OFF_ACC   0                       // 2 floats (+pad to 8)
#define OFF_SOL   8                       // B*16
#define OFF_GRAM  72                      // B*256
#define OFF_DP    1096                    // B*N_
#define OFF_M     (OFF_DP + B_*N_)        // HW_ (per-batch scratch)
#define OFF_G     (OFF_M + HW_)           // HW_
#define OFF_T     (OFF_G + HW_)           // 7*HW_
#define OFF_ONES  (OFF_T + 7*HW_)         // N_ of 1.0f
#define OFF_ZEROS (OFF_ONES + N_)         // N_ of 0.0f
// total ≈ 9.83M floats ≈ 39.3 MB

// ---------------------------------------------------------------------------
__global__ void zero_kernel(float* w) {
    int i = blockIdx.x * blockDim.x + threadIdx.x;
    if (i < OFF_DP) w[i] = 0.f;           // acc + sol + gram regions
}

// constant feature rows so every WMMA lane has a valid pointer (no divergence)
__global__ void fill_rows_kernel(float* ones_row, float* zeros_row) {
    int i = blockIdx.x * blockDim.x + threadIdx.x;
    if (i < N_) { ones_row[i] = 1.f; zeros_row[i] = 0.f; }
}

// 4x4 area downsample of hr_pan -> dp [B, N_]
__global__ void down_kernel(const float* __restrict__ pan, float* __restrict__ dp) {
    int i = blockIdx.x * blockDim.x + threadIdx.x;
    if (i >= B_ * N_) return;
    int b = i / N_, p = i % N_;
    int py = p / ws2_, px = p % ws2_;
    const float* base = pan + (size_t)b * HW_ + (size_t)(py * 4) * W_ + px * 4;
    float s = 0.f;
    #pragma unroll
    for (int r = 0; r < 4; ++r) {
        #pragma unroll
        for (int c = 0; c < 4; ++c) s += base[r * W_ + c];
    }
    dp[i] = s * (1.f / 16.f);
}

// Gram = F F^T per batch via V_WMMA_F32_16X16X4_F32, F rows: [1, lr0..lr7, dp, 0...]
// 64 waves per batch, 1024 pixels per wave, K=4 per WMMA, 4x pipelined.
__global__ void gram_kernel(const float* __restrict__ lr, const float* __restrict__ dp,
                            const float* __restrict__ ones_row,
                            const float* __restrict__ zeros_row,
                            float* __restrict__ gram) {
    const int wave = blockIdx.x * (blockDim.x >> 5) + (threadIdx.x >> 5);
    const int wavesPerBatch = 64;
    const int b   = wave / wavesPerBatch;
    const int wib = wave % wavesPerBatch;
    const int lane = threadIdx.x & 31;
    const int m  = lane & 15;      // feature row (0..15)
    const int kh = lane >> 4;      // K half: 0 -> pixels p,p+1 ; 1 -> p+2,p+3
    const int pixPerWave = N_ / wavesPerBatch;   // 1024
    const int p0 = wib * pixPerWave;

    const float* row;
    if (m == 0)             row = ones_row;
    else if (m <= 8)        row = lr + ((size_t)b * C_ + (m - 1)) * N_;
    else if (m == 9)        row = dp + (size_t)b * N_;
    else                    row = zeros_row;
    const float* rp = row + p0 + 2 * kh;

    v8f acc = {};
    for (int p = 0; p < pixPerWave; p += 16) {
        // 4 loads issued up front, then 4 WMMAs (accumulation chain on C)
        v2f a0 = *(const v2f*)(rp + p);
        v2f a1 = *(const v2f*)(rp + p + 4);
        v2f a2 = *(const v2f*)(rp + p + 8);
        v2f a3 = *(const v2f*)(rp + p + 12);
        // F F^T: identical register image serves as both A (16x4) and B (4x16)
        acc = __builtin_amdgcn_wmma_f32_16x16x4_f32(false, a0, false, a0, (short)0, acc, false, false);
        acc = __builtin_amdgcn_wmma_f32_16x16x4_f32(false, a1, false, a1, (short)0, acc, false, false);
        acc = __builtin_amdgcn_wmma_f32_16x16x4_f32(false, a2, false, a2, (short)0, acc, false, false);
        acc = __builtin_amdgcn_wmma_f32_16x16x4_f32(false, a3, false, a3, (short)0, acc, false, false);
    }
    float* g = gram + b * 256;
    #pragma unroll
    for (int r = 0; r < 8; ++r) {
        const int M = r + 8 * kh;          // C/D layout: VGPR r, lane-half selects M vs M+8
        atomicAdd(&g[M * 16 + m], acc[r]);
    }
}

// Gauss-Jordan with partial pivoting on the 9x9 system; Atb = Gram column 9.
__global__ void solve_kernel(const float* __restrict__ gram, float* __restrict__ sol) {
    int b = threadIdx.x;
    if (b >= B_) return;
    float A[9][10];
    const float* g = gram + b * 256;
    for (int i = 0; i < 9; ++i) {
        for (int j = 0; j < 9; ++j) A[i][j] = g[i * 16 + j];
        A[i][9] = g[i * 16 + 9];
    }
    for (int k = 0; k < 9; ++k) {
        int pr = k; float mx = fabsf(A[k][k]);
        for (int i = k + 1; i < 9; ++i) { float v = fabsf(A[i][k]); if (v > mx) { mx = v; pr = i; } }
        if (pr != k) for (int j = 0; j < 10; ++j) { float t = A[k][j]; A[k][j] = A[pr][j]; A[pr][j] = t; }
        float inv = 1.f / A[k][k];
        for (int j = k; j < 10; ++j) A[k][j] *= inv;
        for (int i = 0; i < 9; ++i) {
            if (i == k) continue;
            float f = A[i][k];
            for (int j = k; j < 10; ++j) A[i][j] -= f * A[k][j];
        }
    }
    for (int i = 0; i < 9; ++i) sol[b * 16 + i] = A[i][9];
}

// M_ and G_ intensity images for batch b
__global__ void mg_kernel(const float* __restrict__ hrm, const float* __restrict__ fus,
                          const float* __restrict__ sol, float* __restrict__ Mb,
                          float* __restrict__ Gb, int b) {
    int q = blockIdx.x * blockDim.x + threadIdx.x;
    const float* s = sol + b * 16;
    float m = s[0], g = s[0];
    #pragma unroll
    for (int c = 0; c < C_; ++c) {
        float w = s[1 + c];
        size_t o = ((size_t)b * C_ + c) * HW_ + q;
        m += w * hrm[o];
        g += w * fus[o];
    }
    Mb[q] = m; Gb[q] = g;
}

__device__ __forceinline__ int reflect_i(int i, int n) {
    i = i < 0 ? -i : i;
    return i >= n ? 2 * n - 2 - i : i;
}

// async 16B copy global -> LDS (gfx1250 ASYNCcnt path)
__device__ __forceinline__ void async_copy_b128(float* lds_dst, const float* gsrc) {
    uint32_t lds = (uint32_t)(uintptr_t)lds_dst;   // low 32 bits of generic LDS ptr == LDS addr
    asm volatile("global_load_async_to_lds_b128 %0, %1, off"
                 :: "v"(lds), "v"(gsrc) : "memory");
}
__device__ __forceinline__ void wait_async_all() {
    asm volatile("s_wait_asynccnt 0x0" ::: "memory");
}

// Horizontal 31-tap blur (reflect) of 7 fields, products formed on the fly.
// Rows staged into LDS with GLOBAL_LOAD_ASYNC_TO_LDS_B128.
__global__ void hblur_kernel(const float* __restrict__ Mb, const float* __restrict__ Gb,
                             const float* __restrict__ pan, float* __restrict__ T, int b) {
    __shared__ __align__(16) float sM[W_];
    __shared__ __align__(16) float sP[W_];
    __shared__ __align__(16) float sG[W_];
    const int y = blockIdx.x;
    const float* prow = pan + (size_t)b * HW_ + (size_t)y * W_;
    const float* mrow = Mb + (size_t)y * W_;
    const float* grow = Gb + (size_t)y * W_;

    const int t4 = threadIdx.x * 4;                // 256 threads x 4 floats = 1024
    async_copy_b128(&sM[t4], mrow + t4);
    async_copy_b128(&sP[t4], prow + t4);
    async_copy_b128(&sG[t4], grow + t4);
    wait_async_all();
    __syncthreads();

    const float inv = 1.f / KBLUR;
    for (int x = threadIdx.x; x < W_; x += blockDim.x) {
        float aM = 0, aP = 0, aMP = 0, aMM = 0, aPP = 0, aG = 0, aGG = 0;
        for (int d = -PADB; d <= PADB; ++d) {
            int xi = reflect_i(x + d, W_);
            float m = sM[xi], p = sP[xi], g = sG[xi];
            aM += m; aP += p; aMP += m * p; aMM += m * m; aPP += p * p; aG += g; aGG += g * g;
        }
        size_t o = (size_t)y * W_ + x;
        T[0 * HW_ + o] = aM * inv;  T[1 * HW_ + o] = aP * inv;
        T[2 * HW_ + o] = aMP * inv; T[3 * HW_ + o] = aMM * inv;
        T[4 * HW_ + o] = aPP * inv; T[5 * HW_ + o] = aG * inv;
        T[6 * HW_ + o] = aGG * inv;
    }
}

// Vertical blur + S / grads / L1 terms + block-reduced accumulation.
__global__ void vfinal_kernel(const float* __restrict__ T, const float* __restrict__ Gb,
                              const float* __restrict__ pan, const float* __restrict__ hrm,
                              const float* __restrict__ fus, float* __restrict__ acc, int b) {
    const int q = blockIdx.x * blockDim.x + threadIdx.x;
    const int y = q / W_, x = q % W_;
    float s0 = 0, s1 = 0, s2 = 0, s3 = 0, s4 = 0, s5 = 0, s6 = 0;
    for (int d = -PADB; d <= PADB; ++d) {
        int yi = reflect_i(y + d, H_);
        size_t o = (size_t)yi * W_ + x;
        s0 += T[0 * HW_ + o]; s1 += T[1 * HW_ + o]; s2 += T[2 * HW_ + o];
        s3 += T[3 * HW_ + o]; s4 += T[4 * HW_ + o]; s5 += T[5 * HW_ + o];
        s6 += T[6 * HW_ + o];
    }
    const float inv = 1.f / KBLUR;
    s0 *= inv; s1 *= inv; s2 *= inv; s3 *= inv; s4 *= inv; s5 *= inv; s6 *= inv;
    float cov  = s2 - s0 * s1;
    float stdM = sqrtf(fabsf(s3 - s0 * s0) + EPS_);
    float stdP = sqrtf(fabsf(s4 - s1 * s1) + EPS_);
    float stdG = sqrtf(fabsf(s6 - s5 * s5) + EPS_);
    float r  = cov / (stdM * stdP);
    float r2 = r * r;
    float S  = r2 * r2;

    float sumc = 0.f;
    #pragma unroll
    for (int c = 0; c < C_; ++c) {
        size_t o = ((size_t)b * C_ + c) * HW_ + q;
        sumc += fabsf(fus[o] - hrm[o]);
    }
    float lc = S * sumc;

    float P = pan[(size_t)b * HW_ + q];
    float gradP = (P - s1) / stdP;
    float gradG = (Gb[q] - s5) / stdG;
    float la = fabsf((gradG - gradP) * (2.f - S));

    __shared__ float rc[256], ra[256];
    rc[threadIdx.x] = lc; ra[threadIdx.x] = la;
    __syncthreads();
    for (int s = 128; s > 0; s >>= 1) {
        if (threadIdx.x < s) { rc[threadIdx.x] += rc[threadIdx.x + s]; ra[threadIdx.x] += ra[threadIdx.x + s]; }
        __syncthreads();
    }
    if (threadIdx.x == 0) { atomicAdd(&acc[0], rc[0]); atomicAdd(&acc[1], ra[0]); }
}

__global__ void final_kernel(const float* __restrict__ acc, float* __restrict__ out) {
    out[0] = acc[0] * (1.f / ((float)B_ * C_ * HW_)) + acc[1] * (1.f / ((float)B_ * HW_));
}

// ---------------------------------------------------------------------------
extern "C" void kernel_launch(void* const* d_in, const int* in_sizes, int n_in,
                              void* d_out, int out_size, void* d_ws, size_t ws_size,
                              hipStream_t stream) {
    const float* lr  = (const float*)d_in[0];   // [B,C,h,w]
    const float* pan = (const float*)d_in[1];   // [B,1,H,W]
    const float* hrm = (const float*)d_in[2];   // [B,C,H,W]
    const float* fus = (const float*)d_in[3];   // [B,C,H,W]
    float* out = (float*)d_out;

    float* Wk    = (float*)d_ws;
    float* acc   = Wk + OFF_ACC;
    float* sol   = Wk + OFF_SOL;
    float* gram  = Wk + OFF_GRAM;
    float* dp    = Wk + OFF_DP;
    float* Mb    = Wk + OFF_M;
    float* Gb    = Wk + OFF_G;
    float* T     = Wk + OFF_T;
    float* ones  = Wk + OFF_ONES;
    float* zeros = Wk + OFF_ZEROS;

    zero_kernel<<<(OFF_DP + 255) / 256, 256, 0, stream>>>(Wk);
    fill_rows_kernel<<<(N_ + 255) / 256, 256, 0, stream>>>(ones, zeros);
    down_kernel<<<(B_ * N_ + 255) / 256, 256, 0, stream>>>(pan, dp);
    gram_kernel<<<B_ * 8, 256, 0, stream>>>(lr, dp, ones, zeros, gram); // 64 waves / batch
    solve_kernel<<<1, 32, 0, stream>>>(gram, sol);

    for (int b = 0; b < B_; ++b) {
        mg_kernel<<<HW_ / 256, 256, 0, stream>>>(hrm, fus, sol, Mb, Gb, b);
        hblur_kernel<<<H_, 256, 0, stream>>>(Mb, Gb, pan, T, b);
        vfinal_kernel<<<HW_ / 256, 256, 0, stream>>>(T, Gb, pan, hrm, fus, acc, b);
    }
    final_kernel<<<1, 1, 0, stream>>>(acc, out);
}